// NetGAT_80135499808835
// MI455X (gfx1250) — compile-verified
//
#include <hip/hip_runtime.h>
#include <hip/hip_bf16.h>

#define NN   50000
#define EE   800000
#define ETOT (EE + NN)
#define CH   128
#define GG   64
#define EPS  1e-5f
#define SLOPE 0.2f
#define XS_STRIDE 132   // 128 + 4: keeps b128 stores 16B-aligned, rotates LDS banks 4/row

typedef __attribute__((ext_vector_type(2))) float v2f;
typedef __attribute__((ext_vector_type(8))) float v8f;
typedef __attribute__((ext_vector_type(4))) int   v4i;

#define AS1 __attribute__((address_space(1)))
#define AS3 __attribute__((address_space(3)))

#if defined(__has_builtin)
#if __has_builtin(__builtin_amdgcn_global_load_async_to_lds_b128)
#define HAVE_ASYNC_LDS 1
#endif
#endif

// ---------- helpers ----------
__device__ __forceinline__ void atomicAddF(float* p, float v) {
    __hip_atomic_fetch_add(p, v, __ATOMIC_RELAXED, __HIP_MEMORY_SCOPE_AGENT);
}
// monotone order-preserving float <-> uint map (for atomicMax-based segment max)
__device__ __forceinline__ unsigned fmap(float x) {
    unsigned u = __float_as_uint(x);
    return (u & 0x80000000u) ? ~u : (u | 0x80000000u);
}
__device__ __forceinline__ float funmap(unsigned u) {
    return __uint_as_float((u & 0x80000000u) ? (u & 0x7FFFFFFFu) : ~u);
}
__device__ __forceinline__ float lrelu(float v) { return v > 0.f ? v : v * SLOPE; }

// ---------- GEMM: H = X @ W  (NxCH = NxCH @ CHxCH), fp32 WMMA ----------
// block = 256 threads = 8 waves; the block's 16-row X tile is staged into LDS once
// (async copy on gfx1250), then wave w computes tile rows[b*16..+16) x cols[w*16..+16).
__global__ void gat_gemm(const float* __restrict__ X, const float* __restrict__ W,
                         float* __restrict__ H) {
    __shared__ float xs[16 * XS_STRIDE];
    const int t = threadIdx.x;
    const int row0 = blockIdx.x * 16;

    // stage 16x128 X tile -> LDS: 512 x b128 chunks, 2 per thread
#pragma unroll
    for (int i = 0; i < 2; ++i) {
        const int c = t + i * 256;
        const int r = c >> 5;               // 0..15
        const int col = (c & 31) * 4;       // 0..124
        const float* g = X + (size_t)(row0 + r) * CH + col;
        float* l = &xs[r * XS_STRIDE + col];
#ifdef HAVE_ASYNC_LDS
        __builtin_amdgcn_global_load_async_to_lds_b128(
            (AS1 v4i*)g, (AS3 v4i*)l, 0, 0);
#else
        *(float4*)l = *(const float4*)g;
#endif
    }
#ifdef HAVE_ASYNC_LDS
    asm volatile("s_wait_asynccnt 0" ::: "memory");
#endif
    __syncthreads();

    const int wave = t >> 5;
    const int lane = t & 31;
    const int col0 = wave * 16;
    const int m  = lane & 15;
    const int kh = lane >> 4;          // 0: K pair {k,k+1}; 1: K pair {k+2,k+3}
    v8f acc = {};
#pragma unroll
    for (int k0 = 0; k0 < CH; k0 += 4) {
        const int ka = k0 + 2 * kh;
        // A fragment from LDS: lane m holds A[m][ka], A[m][ka+1] (bank-conflict-free)
        v2f a = *(const v2f*)&xs[m * XS_STRIDE + ka];
        // B fragment: 4x16 f32 — lane m holds W[ka][col0+m], W[ka+1][col0+m]
        v2f b;
        b.x = W[(size_t)ka * CH + col0 + m];
        b.y = W[(size_t)(ka + 1) * CH + col0 + m];
        acc = __builtin_amdgcn_wmma_f32_16x16x4_f32(false, a, false, b,
                                                    (short)0, acc, false, false);
    }
    // C layout: VGPR r -> M=r (lanes 0-15) / M=r+8 (lanes 16-31), N = m
    const int mbase = row0 + kh * 8;
#pragma unroll
    for (int r = 0; r < 8; ++r)
        H[(size_t)(mbase + r) * CH + col0 + m] = acc[r];
}

// ---------- per-node attention scalars ----------
__global__ void gat_alpha(const float* __restrict__ H, const float* __restrict__ as,
                          const float* __restrict__ ad, float* __restrict__ asrc,
                          float* __restrict__ adst) {
    const int wave = threadIdx.x >> 5;
    const int lane = threadIdx.x & 31;
    const int n = blockIdx.x * 8 + wave;
    if (n >= NN) return;
    float s = 0.f, d = 0.f;
#pragma unroll
    for (int i = lane; i < CH; i += 32) {
        float hv = H[(size_t)n * CH + i];
        s += hv * as[i];
        d += hv * ad[i];
    }
#pragma unroll
    for (int off = 16; off > 0; off >>= 1) {
        s += __shfl_xor(s, off, 32);
        d += __shfl_xor(d, off, 32);
    }
    if (lane == 0) { asrc[n] = s; adst[n] = d; }
}

// ---------- edge pass 1: segment max (mapped-uint atomicMax) ----------
__global__ void edge_max(const int* __restrict__ ei, const float* __restrict__ asrc,
                         const float* __restrict__ adst, unsigned* __restrict__ emax) {
    int e = blockIdx.x * blockDim.x + threadIdx.x;
    if (e >= ETOT) return;
    int s = (e < EE) ? ei[e] : (e - EE);
    int d = (e < EE) ? ei[EE + e] : (e - EE);
    float v = lrelu(asrc[s] + adst[d]);
    atomicMax(&emax[d], fmap(v));
}

// ---------- edge pass 2: softmax denominator ----------
__global__ void edge_denom(const int* __restrict__ ei, const float* __restrict__ asrc,
                           const float* __restrict__ adst, const unsigned* __restrict__ emax,
                           float* __restrict__ denom) {
    int e = blockIdx.x * blockDim.x + threadIdx.x;
    if (e >= ETOT) return;
    int s = (e < EE) ? ei[e] : (e - EE);
    int d = (e < EE) ? ei[EE + e] : (e - EE);
    float v = lrelu(asrc[s] + adst[d]);
    atomicAddF(&denom[d], __expf(v - funmap(emax[d])));
}

// ---------- edge pass 3: weighted scatter-aggregate (one wave per edge) ----------
__global__ void edge_agg(const int* __restrict__ ei, const float* __restrict__ asrc,
                         const float* __restrict__ adst, const unsigned* __restrict__ emax,
                         const float* __restrict__ denom, const float* __restrict__ H,
                         float* __restrict__ AGG) {
    int gid = blockIdx.x * blockDim.x + threadIdx.x;
    int e = gid >> 5;
    if (e >= ETOT) return;
    int lane = gid & 31;
    int s = (e < EE) ? ei[e] : (e - EE);
    int d = (e < EE) ? ei[EE + e] : (e - EE);
    float v = lrelu(asrc[s] + adst[d]);
    float coef = __expf(v - funmap(emax[d])) / denom[d];
    const float* hs = H + (size_t)s * CH + lane * 4;
    float* o = AGG + (size_t)d * CH + lane * 4;
#pragma unroll
    for (int i = 0; i < 4; ++i) atomicAddF(o + i, hs[i] * coef);
}

// ---------- bias add + per-channel sum / sumsq ----------
__global__ void bias_stats(float* __restrict__ Y, const float* __restrict__ b,
                           float* __restrict__ colsum, float* __restrict__ colsumsq) {
    const int c = threadIdx.x;            // 128 threads = 128 channels
    const int r0 = blockIdx.x * 64;
    const int r1 = min(r0 + 64, NN);
    const float bias = b[c];
    float s = 0.f, sq = 0.f;
    for (int r = r0; r < r1; ++r) {
        float y = Y[(size_t)r * CH + c] + bias;
        Y[(size_t)r * CH + c] = y;
        s += y; sq += y * y;
    }
    atomicAddF(&colsum[c], s);
    atomicAddF(&colsumsq[c], sq);
}

// ---------- batch-norm (biased var) + ReLU ----------
__global__ void bn_relu(const float* __restrict__ Y, const float* __restrict__ colsum,
                        const float* __restrict__ colsumsq, const float* __restrict__ gamma,
                        const float* __restrict__ beta, float* __restrict__ Xout) {
    size_t idx = (size_t)blockIdx.x * blockDim.x + threadIdx.x;
    if (idx >= (size_t)NN * CH) return;
    int c = (int)(idx & (CH - 1));
    float mu = colsum[c] * (1.f / NN);
    float var = colsumsq[c] * (1.f / NN) - mu * mu;
    float xn = (Y[idx] - mu) * rsqrtf(var + EPS) * gamma[c] + beta[c];
    Xout[idx] = fmaxf(xn, 0.f);
}

// ---------- graph mean pooling ----------
__global__ void pool_acc(const float* __restrict__ X, const int* __restrict__ batch,
                         float* __restrict__ gsum, float* __restrict__ gcount) {
    const int r = blockIdx.x;             // one block per node, 128 threads
    const int c = threadIdx.x;
    const int g = batch[r];
    atomicAddF(&gsum[(size_t)g * CH + c], X[(size_t)r * CH + c]);
    if (c == 0) atomicAddF(&gcount[g], 1.f);
}
__global__ void pool_div(float* __restrict__ gsum, const float* __restrict__ gcount) {
    int idx = blockIdx.x * blockDim.x + threadIdx.x;
    if (idx >= GG * CH) return;
    gsum[idx] /= fmaxf(gcount[idx / CH], 1.f);
}

extern "C" void kernel_launch(void* const* d_in, const int* in_sizes, int n_in,
                              void* d_out, int out_size, void* d_ws, size_t ws_size,
                              hipStream_t stream) {
    const float* x     = (const float*)d_in[0];
    const int*   ei    = (const int*)d_in[1];
    const int*   batch = (const int*)d_in[2];
    const float* W[3]  = {(const float*)d_in[3],  (const float*)d_in[9],  (const float*)d_in[15]};
    const float* AS[3] = {(const float*)d_in[4],  (const float*)d_in[10], (const float*)d_in[16]};
    const float* AD[3] = {(const float*)d_in[5],  (const float*)d_in[11], (const float*)d_in[17]};
    const float* Bv[3] = {(const float*)d_in[6],  (const float*)d_in[12], (const float*)d_in[18]};
    const float* Ga[3] = {(const float*)d_in[7],  (const float*)d_in[13], (const float*)d_in[19]};
    const float* Be[3] = {(const float*)d_in[8],  (const float*)d_in[14], (const float*)d_in[20]};

    float* ws = (float*)d_ws;
    size_t o = 0;
    float*    Hbuf  = ws + o;             o += (size_t)NN * CH;
    float*    Abuf  = ws + o;             o += (size_t)NN * CH;
    float*    Bbuf  = ws + o;             o += (size_t)NN * CH;
    float*    asrc  = ws + o;             o += NN;
    float*    adst  = ws + o;             o += NN;
    unsigned* emax  = (unsigned*)(ws + o); o += NN;
    float*    denom = ws + o;             o += NN;
    float*    cstat = ws + o;             o += 256;   // colsum[128] | colsumsq[128]
    float*    gcnt  = ws + o;             o += GG;

    float* out  = (float*)d_out;
    float* gsum = out + (size_t)NN * CH;

    (void)hipMemsetAsync(d_out, 0, (size_t)(NN * CH + GG * CH) * sizeof(float), stream);

    const float* Xin = x;
    float* aggs[3] = {Abuf, Bbuf, out};
    for (int l = 0; l < 3; ++l) {
        float* AGG = aggs[l];
        if (l != 2)  // d_out already zeroed
            (void)hipMemsetAsync(AGG, 0, (size_t)NN * CH * sizeof(float), stream);
        (void)hipMemsetAsync(emax,  0, (size_t)NN * sizeof(unsigned), stream);
        (void)hipMemsetAsync(denom, 0, (size_t)NN * sizeof(float), stream);
        (void)hipMemsetAsync(cstat, 0, 256 * sizeof(float), stream);

        gat_gemm <<<NN / 16, 256, 0, stream>>>(Xin, W[l], Hbuf);
        gat_alpha<<<NN / 8,  256, 0, stream>>>(Hbuf, AS[l], AD[l], asrc, adst);
        edge_max  <<<(ETOT + 255) / 256, 256, 0, stream>>>(ei, asrc, adst, emax);
        edge_denom<<<(ETOT + 255) / 256, 256, 0, stream>>>(ei, asrc, adst, emax, denom);
        edge_agg  <<<((size_t)ETOT * 32 + 255) / 256, 256, 0, stream>>>(ei, asrc, adst,
                                                                        emax, denom, Hbuf, AGG);
        bias_stats<<<(NN + 63) / 64, 128, 0, stream>>>(AGG, Bv[l], cstat, cstat + 128);
        bn_relu<<<((size_t)NN * CH + 255) / 256, 256, 0, stream>>>(AGG, cstat, cstat + 128,
                                                                   Ga[l], Be[l], AGG);
        Xin = AGG;
    }

    (void)hipMemsetAsync(gcnt, 0, GG * sizeof(float), stream);
    pool_acc<<<NN, 128, 0, stream>>>(out, batch, gsum, gcnt);
    pool_div<<<(GG * CH + 255) / 256, 256, 0, stream>>>(gsum, gcnt);
}